// S_47330539602539
// MI455X (gfx1250) — compile-verified
//
#include <hip/hip_runtime.h>

// ---------------------------------------------------------------------------
// Trilinear SDF sampling (grid_sample 3D, border clamp, align_corners=True)
// over two 256^3 float volumes with mutually exclusive region masks.
//
// MI455X strategy (memory-bound gather kernel — WMMA is not applicable):
//  * Both volumes (2 x 64 MB) fit in the 192 MB L2 -> keep them resident by
//    streaming coords/output with non-temporal cache hints (CDNA5 TH field).
//  * One sample path per lane via branchless {vol, center, scale} select
//    (wave32 would otherwise pay for both fg and bg paths in every wave).
//  * Pair the two x-neighbors into a single 8-byte gather (4 x b64 per point
//    instead of 8 x b32); border case x0 == 255 handled by lane select.
// ---------------------------------------------------------------------------

typedef float v2f __attribute__((ext_vector_type(2), aligned(4)));

#define GRID    256
#define ROW_B   (GRID * 4)           // bytes per W-row
#define PLANE_B (GRID * GRID * 4)    // bytes per (H,W) plane

__global__ __launch_bounds__(256) void sdf_sample_kernel(
    const float* __restrict__ x,     // [B*N, 3]
    const float* __restrict__ fg,    // [256,256,256]
    const float* __restrict__ bg,    // [256,256,256]
    const float* __restrict__ fgc,   // [3]
    const float* __restrict__ fgs,   // [3]
    const float* __restrict__ bgc,   // [3]
    const float* __restrict__ bgs,   // [3]
    float* __restrict__ out,         // [B*N]
    int n)
{
    int i = blockIdx.x * blockDim.x + threadIdx.x;
    if (i >= n) return;

    // --- streaming coordinate load: non-temporal (don't pollute L2) --------
    const float* pp = x + (size_t)3 * (size_t)i;
    float p0 = __builtin_nontemporal_load(pp + 0);
    float p1 = __builtin_nontemporal_load(pp + 1);
    float p2 = __builtin_nontemporal_load(pp + 2);

    // --- region masks (strict inequalities, fg excludes bg) ----------------
    bool is_f = (p0 > -1.0f) && (p0 < 1.0f) &&
                (p1 > -1.0f) && (p1 < 1.0f) &&
                (p2 > -1.0f) && (p2 < 1.0f);
    bool is_b = (p0 > -4.0f) && (p0 < 4.0f) &&
                (p1 > -4.0f) && (p1 < 4.0f) &&
                (p2 > -4.0f) && (p2 < 4.0f);
    bool valid = is_f || is_b;

    // --- uniform params (kernarg-relative -> scalar s_load path) -----------
    float fc0 = fgc[0], fc1 = fgc[1], fc2 = fgc[2];
    float fs0 = fgs[0], fs1 = fgs[1], fs2 = fgs[2];
    float bc0 = bgc[0], bc1 = bgc[1], bc2 = bgc[2];
    float bs0 = bgs[0], bs1 = bgs[1], bs2 = bgs[2];

    // --- branchless per-lane volume/param select ---------------------------
    const float* vol = is_f ? fg : bg;
    float c0 = is_f ? fc0 : bc0;
    float c1 = is_f ? fc1 : bc1;
    float c2 = is_f ? fc2 : bc2;
    float s0 = is_f ? fs0 : bs0;
    float s1 = is_f ? fs1 : bs1;
    float s2 = is_f ? fs2 : bs2;

    // normalize + reverse (xyz -> zyx): W index <- p2, H <- p1, D <- p0
    float cx = (p2 - c2) / s2;
    float cy = (p1 - c1) / s1;
    float cz = (p0 - c0) / s0;

    // align_corners=True unnormalize + border clamp: (c+1)*0.5*(G-1)
    float fx = fminf(fmaxf(fmaf(cx, 127.5f, 127.5f), 0.0f), 255.0f);
    float fy = fminf(fmaxf(fmaf(cy, 127.5f, 127.5f), 0.0f), 255.0f);
    float fz = fminf(fmaxf(fmaf(cz, 127.5f, 127.5f), 0.0f), 255.0f);

    float x0f = floorf(fx); int X0 = (int)x0f; float wx = fx - x0f;
    float y0f = floorf(fy); int Y0 = (int)y0f; float wy = fy - y0f;
    float z0f = floorf(fz); int Z0 = (int)z0f; float wz = fz - z0f;

    int XB = min(X0, GRID - 2);       // base for the paired x-load
    int Y1 = min(Y0 + 1, GRID - 1);
    int Z1 = min(Z0 + 1, GRID - 1);

    // --- 4 paired gathers (b64 each), issued back-to-back ------------------
    const char* base = (const char*)vol;
    unsigned zo0 = (unsigned)Z0 * (unsigned)PLANE_B;
    unsigned zo1 = (unsigned)Z1 * (unsigned)PLANE_B;
    unsigned yo0 = (unsigned)Y0 * (unsigned)ROW_B;
    unsigned yo1 = (unsigned)Y1 * (unsigned)ROW_B;
    unsigned xo  = (unsigned)XB * 4u;

    v2f v00 = *(const v2f*)(base + (zo0 + yo0 + xo));  // (z0, y0, xb..xb+1)
    v2f v01 = *(const v2f*)(base + (zo0 + yo1 + xo));  // (z0, y1, ...)
    v2f v10 = *(const v2f*)(base + (zo1 + yo0 + xo));  // (z1, y0, ...)
    v2f v11 = *(const v2f*)(base + (zo1 + yo1 + xo));  // (z1, y1, ...)

    // border fix: if X0 == 255 then XB = 254 and both taps are element .y
    // (wx == 0 there, so c001 is irrelevant anyway)
    bool hi = X0 > XB;
    float c000 = hi ? v00.y : v00.x, c001 = v00.y;
    float c010 = hi ? v01.y : v01.x, c011 = v01.y;
    float c100 = hi ? v10.y : v10.x, c101 = v10.y;
    float c110 = hi ? v11.y : v11.x, c111 = v11.y;

    // --- trilinear blend (same formulation as reference) -------------------
    float iwx = 1.0f - wx;
    float c00 = c000 * iwx + c001 * wx;
    float c01 = c010 * iwx + c011 * wx;
    float c10 = c100 * iwx + c101 * wx;
    float c11 = c110 * iwx + c111 * wx;
    float iwy = 1.0f - wy;
    float d0 = c00 * iwy + c01 * wy;
    float d1 = c10 * iwy + c11 * wy;
    float r  = d0 * (1.0f - wz) + d1 * wz;

    r = valid ? r : 1.0f;

    // streaming output: non-temporal store (keep L2 for the volumes)
    __builtin_nontemporal_store(r, out + i);
}

extern "C" void kernel_launch(void* const* d_in, const int* in_sizes, int n_in,
                              void* d_out, int out_size, void* d_ws, size_t ws_size,
                              hipStream_t stream) {
    const float* x_i  = (const float*)d_in[0];  // [B*N*3]
    const float* fg   = (const float*)d_in[1];  // [256^3]
    const float* bg   = (const float*)d_in[2];  // [256^3]
    const float* fgc  = (const float*)d_in[3];  // [3]
    const float* fgs  = (const float*)d_in[4];  // [3]
    const float* bgc  = (const float*)d_in[5];  // [3]
    const float* bgs  = (const float*)d_in[6];  // [3]
    float* out = (float*)d_out;

    int n = in_sizes[0] / 3;                    // B*N points
    int block = 256;                            // 8 wave32 per workgroup
    int grid  = (n + block - 1) / block;

    sdf_sample_kernel<<<grid, block, 0, stream>>>(x_i, fg, bg, fgc, fgs,
                                                  bgc, bgs, out, n);
}